// Node2District_89206470738329
// MI455X (gfx1250) — compile-verified
//
#include <hip/hip_runtime.h>

#define DIM_IN   128
#define DIM_HID  256
#define DIM_OUT  128
#define N_DIST   1024
#define N_NODES  500000

typedef float v2f __attribute__((ext_vector_type(2)));
typedef float v4f __attribute__((ext_vector_type(4)));
typedef float v8f __attribute__((ext_vector_type(8)));

// ---------------------------------------------------------------------------
// Kernel 0: zero the district accumulator (512 KB in d_ws) every launch.
// ---------------------------------------------------------------------------
__global__ void n2d_zero_head(float* __restrict__ head) {
    int i = blockIdx.x * blockDim.x + threadIdx.x;
    const int n = N_DIST * DIM_IN;
    for (; i < n; i += gridDim.x * blockDim.x) head[i] = 0.0f;
}

// ---------------------------------------------------------------------------
// Kernel 1: segment sum. One wave per node row: 32 lanes x float4 = 128 f32.
// x is streamed with non-temporal 128-bit loads (read-once, 256 MB — keep it
// out of L2); head (512 KB) stays L2-resident and absorbs the f32 atomics
// (no-return global_atomic_add_f32, STOREcnt-tracked, fully pipelined).
// ---------------------------------------------------------------------------
__global__ void n2d_segsum(const float* __restrict__ x,
                           const int*   __restrict__ zone,
                           float*       __restrict__ head) {
    const int lane   = threadIdx.x & 31;
    const int wave   = (blockIdx.x * blockDim.x + threadIdx.x) >> 5;
    const int nwaves = (gridDim.x * blockDim.x) >> 5;

    for (int node = wave; node < N_NODES; node += nwaves) {
        const int z = zone[node];                      // uniform across wave
        const v4f* p = (const v4f*)(x + (size_t)node * DIM_IN) + lane;
        v4f v = __builtin_nontemporal_load(p);         // global_load_b128, TH=NT
        float* h = head + z * DIM_IN + lane * 4;
        atomicAdd(h + 0, v.x);                         // global_atomic_add_f32
        atomicAdd(h + 1, v.y);
        atomicAdd(h + 2, v.z);
        atomicAdd(h + 3, v.w);
    }
}

// ---------------------------------------------------------------------------
// Kernel 2: fused MLP on [1024 x 128] via V_WMMA_F32_16X16X4_F32 (exact f32).
// One workgroup (8 waves) per 16-row district band.
//   Stage 1: h1[16x256] = relu(relu(head_tile) @ w1 + b1)  -> LDS (16 KB)
//   Stage 2: out_tile   = h1 @ w2 + b2                     -> global
// A-operand layout (16x4 f32): lane<16 holds K=k,k+1; lane>=16 holds K=k+2,k+3
// of row (lane&15). B-operand (4x16): same K split, N = lane&15.
// C/D (16x16 f32, 8 VGPRs): VGPR r -> M = r + 8*(lane>=16), N = lane&15.
// ---------------------------------------------------------------------------
__global__ void n2d_mlp(const float* __restrict__ head,
                        const float* __restrict__ w1, const float* __restrict__ b1,
                        const float* __restrict__ w2, const float* __restrict__ b2,
                        float* __restrict__ out) {
    __shared__ float h1[16 * DIM_HID];

    const int lane = threadIdx.x & 31;
    const int wv   = threadIdx.x >> 5;   // wave id 0..7
    const int half = lane >> 4;          // 0: K lo pair, 1: K hi pair
    const int m    = lane & 15;          // row (A) / col (B,C,D) within tile
    const int row0 = blockIdx.x * 16;

    // ---- Stage 1: hidden tile, 16 N-tiles of 16 over 8 waves ----
    for (int t = wv; t < DIM_HID / 16; t += 8) {
        const int n0 = t * 16;
        v8f acc = {};
        for (int k = 0; k < DIM_IN; k += 4) {
            const int ka = k + half * 2;
            v2f a, b;
            float a0 = head[(row0 + m) * DIM_IN + ka];
            float a1 = head[(row0 + m) * DIM_IN + ka + 1];
            a.x = a0 > 0.0f ? a0 : 0.0f;               // relu(head)
            a.y = a1 > 0.0f ? a1 : 0.0f;
            b.x = w1[(size_t)ka       * DIM_HID + n0 + m];
            b.y = w1[(size_t)(ka + 1) * DIM_HID + n0 + m];
            acc = __builtin_amdgcn_wmma_f32_16x16x4_f32(
                      false, a, false, b, (short)0, acc, false, false);
        }
        const float bias = b1[n0 + m];
        #pragma unroll
        for (int r = 0; r < 8; ++r) {
            float v = acc[r] + bias;
            h1[(r + half * 8) * DIM_HID + n0 + m] = v > 0.0f ? v : 0.0f;
        }
    }
    __syncthreads();

    // ---- Stage 2: output tile, 8 N-tiles of 16, one per wave ----
    {
        const int n0 = wv * 16;
        v8f acc = {};
        for (int k = 0; k < DIM_HID; k += 4) {
            const int ka = k + half * 2;
            v2f a, b;
            a.x = h1[m * DIM_HID + ka];
            a.y = h1[m * DIM_HID + ka + 1];
            b.x = w2[(size_t)ka       * DIM_OUT + n0 + m];
            b.y = w2[(size_t)(ka + 1) * DIM_OUT + n0 + m];
            acc = __builtin_amdgcn_wmma_f32_16x16x4_f32(
                      false, a, false, b, (short)0, acc, false, false);
        }
        const float bias = b2[n0 + m];
        #pragma unroll
        for (int r = 0; r < 8; ++r)
            out[(size_t)(row0 + r + half * 8) * DIM_OUT + n0 + m] = acc[r] + bias;
    }
}

// ---------------------------------------------------------------------------
extern "C" void kernel_launch(void* const* d_in, const int* in_sizes, int n_in,
                              void* d_out, int out_size, void* d_ws, size_t ws_size,
                              hipStream_t stream) {
    const float* x    = (const float*)d_in[0];
    const int*   zone = (const int*)  d_in[1];
    const float* w1   = (const float*)d_in[2];
    const float* b1   = (const float*)d_in[3];
    const float* w2   = (const float*)d_in[4];
    const float* b2   = (const float*)d_in[5];

    float* head = (float*)d_ws;        // 1024*128 f32 = 512 KB scratch
    float* out  = (float*)d_out;       // 1024*128 f32

    n2d_zero_head<<<256, 256, 0, stream>>>(head);
    // 2048 blocks * 256 thr = 16384 waves, ~31 node rows each: saturates HBM.
    n2d_segsum<<<2048, 256, 0, stream>>>(x, zone, head);
    // 64 workgroups: one 16-row district band each, 8 waves per workgroup.
    n2d_mlp<<<N_DIST / 16, 256, 0, stream>>>(head, w1, b1, w2, b2, out);
}